// QLayer_65481071399192
// MI455X (gfx1250) — compile-verified
//
#include <hip/hip_runtime.h>
#include <hip/hip_bf16.h>
#include <stdint.h>

// QLayer(16 wires, batch 512) closed form:
//   c_j = cos(x[b,j] + theta[j])
//   out[b,0]   = prod_{j=1..15} c_j
//   out[b,w>0] = prod_{j=0..w}  c_j
// RX layer on |0> -> product state with <Z_j> = cos(a_j); the CNOT ring is a
// classical basis permutation whose final bit on wire w is the prefix-XOR
// b0^..^bw (wire 0: b1^..^b15); E[(-1)^XOR] over independent bits factorizes
// into the product of cosines.

#define NW    16
#define BATCH 512

// Branchless cos for |a| < ~100: Cody-Waite reduction mod pi/2 + minimax
// sin/cos polynomials on [-pi/4, pi/4]. ~1-2 ulp over our input range.
__device__ __forceinline__ float cos_bounded(float a) {
    const float TWO_OVER_PI = 0.636619772367581343f;
    const float PIO2_H = 1.57079637050628662109375f;   // 0x1.921fb6p0
    const float PIO2_M = -4.37113900018624283e-8f;     // pi/2 - PIO2_H
    float fk = rintf(a * TWO_OVER_PI);
    int   k  = (int)fk;
    float r  = fmaf(fk, -PIO2_H, a);
    r        = fmaf(fk, -PIO2_M, r);
    float r2 = r * r;
    // sin(r) = r + r^3 * sp(r^2)
    float sp = fmaf(r2, fmaf(r2, 2.86567956e-6f, -1.98559923e-4f), 8.33338592e-3f);
    sp       = fmaf(r2, sp, -1.66666672e-1f);
    float s  = fmaf(r * r2, sp, r);
    // cos(r) = 1 + r^2 * cp(r^2)
    float cp = fmaf(r2, fmaf(r2, 2.44677067e-5f, -1.38877297e-3f), 4.16666567e-2f);
    cp       = fmaf(r2, cp, -0.5f);
    float c  = fmaf(r2, cp, 1.0f);
    // k%4: 0 -> +cos, 1 -> -sin, 2 -> -cos, 3 -> +sin
    float v  = (k & 1) ? s : c;
    return (((k + 1) & 2) != 0) ? -v : v;
}

__global__ __launch_bounds__(256) void qlayer_expval_kernel(
    const float* __restrict__ x,      // [BATCH, NW]
    const float* __restrict__ theta,  // [NW]
    float* __restrict__ out)          // [BATCH, NW]
{
    __shared__ float s_theta[NW];
    const int t = threadIdx.x;

    // Stage theta (64B) into LDS via CDNA5 async global->LDS DMA (ASYNCcnt).
    if (t < 4) {
        unsigned lds_off = (unsigned)(uintptr_t)(s_theta) + (unsigned)(t * 16);
        const float* g = theta + t * 4;
        asm volatile("global_load_async_to_lds_b128 %0, %1, off"
                     :: "v"(lds_off), "v"(g)
                     : "memory");
    }
    asm volatile("s_wait_asynccnt 0" ::: "memory");
    __syncthreads();

    const int row = blockIdx.x * blockDim.x + t;   // grid covers exactly BATCH

    // 64B vector loads of this row of x.
    float xv[NW];
    const float4* xr = (const float4*)(x + row * NW);
    ((float4*)xv)[0] = xr[0];
    ((float4*)xv)[1] = xr[1];
    ((float4*)xv)[2] = xr[2];
    ((float4*)xv)[3] = xr[3];

    float c[NW];
#pragma unroll
    for (int j = 0; j < NW; ++j) {
        c[j] = cos_bounded(xv[j] + s_theta[j]);
    }

    float o[NW];
    // Prefix products for wires 1..15.
    float p = c[0];
#pragma unroll
    for (int w = 1; w < NW; ++w) {
        p *= c[w];
        o[w] = p;
    }
    // Wire 0: product over j = 1..15 (no division -> safe when c[0] ~ 0).
    float q = c[1];
#pragma unroll
    for (int j = 2; j < NW; ++j) q *= c[j];
    o[0] = q;

    float4* orow = (float4*)(out + row * NW);
    orow[0] = ((float4*)o)[0];
    orow[1] = ((float4*)o)[1];
    orow[2] = ((float4*)o)[2];
    orow[3] = ((float4*)o)[3];
}

extern "C" void kernel_launch(void* const* d_in, const int* in_sizes, int n_in,
                              void* d_out, int out_size, void* d_ws, size_t ws_size,
                              hipStream_t stream) {
    (void)in_sizes; (void)n_in; (void)d_ws; (void)ws_size; (void)out_size;
    const float* x     = (const float*)d_in[0];   // [512,16] f32
    const float* theta = (const float*)d_in[1];   // [16] f32
    float* out         = (float*)d_out;           // [512,16] f32

    qlayer_expval_kernel<<<BATCH / 256, 256, 0, stream>>>(x, theta, out);
}